// MambaBlock3D_85933705658394
// MI455X (gfx1250) — compile-verified
//
#include <hip/hip_runtime.h>
#include <hip/hip_bf16.h>
#include <math.h>

#define DIM      384
#define D_STATE  16
#define D_CONV   4
#define D_INNER  768
#define DT_RANK  24
#define LSEQ     2048
#define BATCH    2
#define BL       (BATCH * LSEQ)        // 4096 rows
#define NXZ      (2 * D_INNER)         // 1536
#define NDBL     (DT_RANK + 2*D_STATE) // 56
#define NDBL_PAD 64

typedef __attribute__((ext_vector_type(2))) float v2f;
typedef __attribute__((ext_vector_type(8))) float v8f;

// CDNA5 fp32 WMMA: D(16x16,f32) = A(16x4,f32) * B(4x16,f32) + C
__device__ __forceinline__ v8f wmma4(v2f a, v2f b, v8f c) {
  return __builtin_amdgcn_wmma_f32_16x16x4_f32(
      /*neg_a=*/false, a, /*neg_b=*/false, b,
      /*c_mod=*/(short)0, c, /*reuse_a=*/false, /*reuse_b=*/false);
}

// One wave accumulates a 16x16 tile of A(MxK,lda) * B(KxN,ldb).
// Columns of B at n >= nvalid read as zero (select, not branch: EXEC stays full).
__device__ __forceinline__ v8f gemm_tile(const float* __restrict__ A, int lda,
                                         const float* __restrict__ B, int ldb,
                                         int K, int tm, int tn, int nvalid) {
  const int lane = threadIdx.x & 31;
  const int half = lane >> 4;      // 0: K lo pair, 1: K hi pair
  const int l15  = lane & 15;
  const int m = tm * 16 + l15;
  const int n = tn * 16 + l15;
  const bool nok = (n < nvalid);
  const float* arow = A + (size_t)m * lda;
  v8f acc = {0.f, 0.f, 0.f, 0.f, 0.f, 0.f, 0.f, 0.f};
  for (int k = 0; k < K; k += 4) {
    const int ka = k + half * 2;
    v2f a;
    a.x = arow[ka];
    a.y = arow[ka + 1];          // contiguous -> b64 load
    v2f b;
    b.x = nok ? B[(size_t)ka * ldb + n] : 0.f;
    b.y = nok ? B[(size_t)(ka + 1) * ldb + n] : 0.f;
    acc = wmma4(a, b, acc);
  }
  return acc;
}

__device__ __forceinline__ float silu_f(float v) { return v / (1.f + expf(-v)); }
__device__ __forceinline__ float softplus_f(float v) {
  return v > 20.f ? v : log1pf(expf(v));
}

// ---------------- LayerNorm over C, gather transposed input ----------------
// one wave per (b,l) position; x is (B, C, L) with stride L over channels
__global__ void k_layernorm(const float* __restrict__ x,
                            const float* __restrict__ ln_w,
                            const float* __restrict__ ln_b,
                            float* __restrict__ xn) {
  const int wid = (blockIdx.x * blockDim.x + threadIdx.x) >> 5;
  if (wid >= BL) return;
  const int lane = threadIdx.x & 31;
  const int b = wid >> 11, l = wid & (LSEQ - 1);
  const float* col = x + (size_t)b * DIM * LSEQ + l;
  float v[DIM / 32];
  float s = 0.f;
#pragma unroll
  for (int i = 0; i < DIM / 32; ++i) {
    v[i] = col[(size_t)(lane + 32 * i) * LSEQ];
    s += v[i];
  }
#pragma unroll
  for (int off = 16; off >= 1; off >>= 1) s += __shfl_xor(s, off, 32);
  const float mu = s * (1.f / DIM);
  float q = 0.f;
#pragma unroll
  for (int i = 0; i < DIM / 32; ++i) { float d = v[i] - mu; q += d * d; }
#pragma unroll
  for (int off = 16; off >= 1; off >>= 1) q += __shfl_xor(q, off, 32);
  const float inv = rsqrtf(q * (1.f / DIM) + 1e-5f);
#pragma unroll
  for (int i = 0; i < DIM / 32; ++i) {
    const int c = lane + 32 * i;
    xn[(size_t)wid * DIM + c] = (v[i] - mu) * inv * ln_w[c] + ln_b[c];
  }
}

// ---------------- GEMM1: xz = xn(4096x384) @ W_in(384x1536) ----------------
__global__ void k_gemm_in(const float* __restrict__ xn,
                          const float* __restrict__ W_in,
                          float* __restrict__ xz) {
  const int wid = (blockIdx.x * blockDim.x + threadIdx.x) >> 5;
  const int ntn = NXZ / 16;                       // 96
  const int tm = wid / ntn, tn = wid % ntn;
  if (tm >= BL / 16) return;
  v8f acc = gemm_tile(xn, DIM, W_in, NXZ, DIM, tm, tn, NXZ);
  const int lane = threadIdx.x & 31, half = lane >> 4, l15 = lane & 15;
  const int n = tn * 16 + l15;
#pragma unroll
  for (int r = 0; r < 8; ++r) {
    const int m = tm * 16 + half * 8 + r;
    xz[(size_t)m * NXZ + n] = acc[r];
  }
}

// ---------------- causal depthwise conv(4) + SiLU -> xc --------------------
__global__ void k_conv_silu(const float* __restrict__ xz,
                            const float* __restrict__ conv_w,
                            const float* __restrict__ conv_b,
                            float* __restrict__ xc) {
  const int idx = blockIdx.x * blockDim.x + threadIdx.x;
  if (idx >= BL * D_INNER) return;
  const int d = idx % D_INNER;
  const int bl = idx / D_INNER;
  const int b = bl >> 11, l = bl & (LSEQ - 1);
  float acc = conv_b[d];
#pragma unroll
  for (int k = 0; k < D_CONV; ++k) {
    const int ll = l + k - (D_CONV - 1);
    if (ll >= 0) acc += xz[(size_t)(b * LSEQ + ll) * NXZ + d] * conv_w[d * D_CONV + k];
  }
  xc[(size_t)bl * D_INNER + d] = silu_f(acc);
}

// ---------------- GEMM2: dbl = xc(4096x768) @ W_xproj(768x56) --------------
__global__ void k_gemm_xproj(const float* __restrict__ xc,
                             const float* __restrict__ W_xproj,
                             float* __restrict__ dbl) {
  const int wid = (blockIdx.x * blockDim.x + threadIdx.x) >> 5;
  const int ntn = NDBL_PAD / 16;                  // 4
  const int tm = wid / ntn, tn = wid % ntn;
  if (tm >= BL / 16) return;
  v8f acc = gemm_tile(xc, D_INNER, W_xproj, NDBL, D_INNER, tm, tn, NDBL);
  const int lane = threadIdx.x & 31, half = lane >> 4, l15 = lane & 15;
  const int n = tn * 16 + l15;
  if (n < NDBL) {
#pragma unroll
    for (int r = 0; r < 8; ++r) {
      const int m = tm * 16 + half * 8 + r;
      dbl[(size_t)m * NDBL + n] = acc[r];
    }
  }
}

// ------- GEMM3: dt = softplus(dbl[:, :24](4096x24) @ W_dt(24x768) + b_dt) --
__global__ void k_gemm_dt(const float* __restrict__ dbl,
                          const float* __restrict__ W_dt,
                          const float* __restrict__ b_dt,
                          float* __restrict__ dt) {
  const int wid = (blockIdx.x * blockDim.x + threadIdx.x) >> 5;
  const int ntn = D_INNER / 16;                   // 48
  const int tm = wid / ntn, tn = wid % ntn;
  if (tm >= BL / 16) return;
  // A has row stride NDBL(56) but only K=24 columns are consumed (6 WMMA steps)
  v8f acc = gemm_tile(dbl, NDBL, W_dt, D_INNER, DT_RANK, tm, tn, D_INNER);
  const int lane = threadIdx.x & 31, half = lane >> 4, l15 = lane & 15;
  const int n = tn * 16 + l15;
  const float bias = b_dt[n];
#pragma unroll
  for (int r = 0; r < 8; ++r) {
    const int m = tm * 16 + half * 8 + r;
    dt[(size_t)m * D_INNER + n] = softplus_f(acc[r] + bias);
  }
}

// ---------------- selective scan over L (the sequential core) --------------
// half-wave (16 lanes == 16 states) per (b,d); fused gate epilogue
__global__ void k_scan(const float* __restrict__ dt,
                       const float* __restrict__ xc,
                       const float* __restrict__ dbl,
                       const float* __restrict__ xz,
                       const float* __restrict__ A_log,
                       const float* __restrict__ D_param,
                       float* __restrict__ y) {
  const int tid = blockIdx.x * blockDim.x + threadIdx.x;
  const int grp = tid >> 4;                        // (b,d) group
  if (grp >= BATCH * D_INNER) return;
  const int n = tid & 15;
  const int b = grp / D_INNER;
  const int d = grp % D_INNER;
  const float An = -expf(A_log[d * D_STATE + n]);
  const float Dp = D_param[d];
  float h = 0.f;
  const size_t base = (size_t)b * LSEQ;
  for (int l = 0; l < LSEQ; ++l) {
    const size_t row = base + l;
    const float dtv = dt[row * D_INNER + d];
    const float xcv = xc[row * D_INNER + d];
    const float Bn  = dbl[row * NDBL + DT_RANK + n];
    const float Cn  = dbl[row * NDBL + DT_RANK + D_STATE + n];
    h = expf(dtv * An) * h + dtv * xcv * Bn;
    float p = h * Cn;                              // reduce over 16 states
    p += __shfl_xor(p, 1, 32);
    p += __shfl_xor(p, 2, 32);
    p += __shfl_xor(p, 4, 32);
    p += __shfl_xor(p, 8, 32);
    if (n == 0) {
      const float zv = xz[row * NXZ + D_INNER + d];
      y[row * D_INNER + d] = (p + xcv * Dp) * silu_f(zv);
    }
  }
}

// ---- GEMM4: out = y(4096x768) @ W_out(768x384), +residual, store (B,C,L) --
__global__ void k_gemm_out(const float* __restrict__ y,
                           const float* __restrict__ W_out,
                           const float* __restrict__ x,
                           float* __restrict__ out) {
  const int wid = (blockIdx.x * blockDim.x + threadIdx.x) >> 5;
  const int ntn = DIM / 16;                        // 24
  const int tm = wid / ntn, tn = wid % ntn;
  if (tm >= BL / 16) return;
  v8f acc = gemm_tile(y, D_INNER, W_out, DIM, D_INNER, tm, tn, DIM);
  const int lane = threadIdx.x & 31, half = lane >> 4, l15 = lane & 15;
  const int n = tn * 16 + l15;                     // channel c
#pragma unroll
  for (int r = 0; r < 8; ++r) {
    const int m = tm * 16 + half * 8 + r;          // global row = b*L + l
    const int b = m >> 11, l = m & (LSEQ - 1);
    const size_t oi = ((size_t)(b * DIM + n)) * LSEQ + l;
    out[oi] = x[oi] + acc[r];
  }
}

extern "C" void kernel_launch(void* const* d_in, const int* in_sizes, int n_in,
                              void* d_out, int out_size, void* d_ws, size_t ws_size,
                              hipStream_t stream) {
  const float* x       = (const float*)d_in[0];
  const float* ln_w    = (const float*)d_in[1];
  const float* ln_b    = (const float*)d_in[2];
  const float* W_in    = (const float*)d_in[3];
  const float* conv_w  = (const float*)d_in[4];
  const float* conv_b  = (const float*)d_in[5];
  const float* W_xproj = (const float*)d_in[6];
  const float* W_dt    = (const float*)d_in[7];
  const float* b_dt    = (const float*)d_in[8];
  const float* A_log   = (const float*)d_in[9];
  const float* D_param = (const float*)d_in[10];
  const float* W_out   = (const float*)d_in[11];
  float* out = (float*)d_out;

  // workspace layout (fp32)
  float* xn  = (float*)d_ws;                    // 4096 x 384
  float* xz  = xn  + (size_t)BL * DIM;          // 4096 x 1536
  float* xc  = xz  + (size_t)BL * NXZ;          // 4096 x 768
  float* dbl = xc  + (size_t)BL * D_INNER;      // 4096 x 56
  float* dtb = dbl + (size_t)BL * NDBL;         // 4096 x 768
  float* yb  = dtb + (size_t)BL * D_INNER;      // 4096 x 768

  const int TPB = 128;                          // 4 wave32 waves

  // 1. LayerNorm: 4096 waves
  k_layernorm<<<(BL * 32) / TPB, TPB, 0, stream>>>(x, ln_w, ln_b, xn);

  // 2. GEMM in: (4096/16)*(1536/16) = 24576 tiles/waves
  k_gemm_in<<<(24576 * 32) / TPB, TPB, 0, stream>>>(xn, W_in, xz);

  // 3. conv + silu: 4096*768 threads
  k_conv_silu<<<(BL * D_INNER) / 256, 256, 0, stream>>>(xz, conv_w, conv_b, xc);

  // 4. GEMM xproj: 256*4 = 1024 tiles
  k_gemm_xproj<<<(1024 * 32) / TPB, TPB, 0, stream>>>(xc, W_xproj, dbl);

  // 5. GEMM dt + softplus: 256*48 = 12288 tiles
  k_gemm_dt<<<(12288 * 32) / TPB, TPB, 0, stream>>>(dbl, W_dt, b_dt, dtb);

  // 6. scan: 1536 half-wave groups -> 24576 threads
  k_scan<<<(BATCH * D_INNER * 16) / TPB, TPB, 0, stream>>>(dtb, xc, dbl, xz,
                                                           A_log, D_param, yb);

  // 7. GEMM out + residual: 256*24 = 6144 tiles
  k_gemm_out<<<(6144 * 32) / TPB, TPB, 0, stream>>>(yb, W_out, x, out);
}